// MyModel_53231824667138
// MI455X (gfx1250) — compile-verified
//
#include <hip/hip_runtime.h>

typedef __attribute__((ext_vector_type(2))) float v2f;
typedef __attribute__((ext_vector_type(8))) float v8f;

#define HH 768
#define BB 256
#define CC 200

// ---------------------------------------------------------------------------
// Kernel 1: fused cosine-mask + masked sums.
// One block per batch row b. 8 waves; each wave handles comments c = wave+8i.
// Per lane: 24 H-columns as 6 float4s. dot>0 decides +/- bucket (T==0 so the
// norms cancel). Register accumulators, combined across waves via ds_add_f32.
// ---------------------------------------------------------------------------
__global__ __launch_bounds__(256)
void mask_sum_kernel(const float* __restrict__ tweet, const float* __restrict__ cemb,
                     float* __restrict__ sum_p, float* __restrict__ sum_m,
                     float* __restrict__ cnt_p, float* __restrict__ cnt_m)
{
    const int b    = blockIdx.x;
    const int tid  = threadIdx.x;
    const int wave = tid >> 5;
    const int lane = tid & 31;

    __shared__ float s_p[HH];
    __shared__ float s_m[HH];
    __shared__ int   s_cp, s_cm;
    for (int h = tid; h < HH; h += 256) { s_p[h] = 0.f; s_m[h] = 0.f; }
    if (tid == 0) { s_cp = 0; s_cm = 0; }
    __syncthreads();

    const float4* trow = reinterpret_cast<const float4*>(tweet + (size_t)b * HH);
    float4 tw[6];
    #pragma unroll
    for (int j = 0; j < 6; ++j) tw[j] = trow[lane + 32 * j];

    float4 ap[6], am[6];
    #pragma unroll
    for (int j = 0; j < 6; ++j) {
        ap[j] = make_float4(0.f, 0.f, 0.f, 0.f);
        am[j] = make_float4(0.f, 0.f, 0.f, 0.f);
    }
    int cp = 0, cm = 0;

    for (int c = wave; c < CC; c += 8) {
        const float4* row =
            reinterpret_cast<const float4*>(cemb + ((size_t)b * CC + c) * HH);
        float4 v[6];
        float  d = 0.f;
        #pragma unroll
        for (int j = 0; j < 6; ++j) {
            v[j] = row[lane + 32 * j];
            d += v[j].x * tw[j].x + v[j].y * tw[j].y +
                 v[j].z * tw[j].z + v[j].w * tw[j].w;
        }
        #pragma unroll
        for (int off = 16; off >= 1; off >>= 1) d += __shfl_xor(d, off, 32);

        if (d > 0.f) {   // wave-uniform branch
            #pragma unroll
            for (int j = 0; j < 6; ++j) {
                ap[j].x += v[j].x; ap[j].y += v[j].y;
                ap[j].z += v[j].z; ap[j].w += v[j].w;
            }
            ++cp;
        } else {
            #pragma unroll
            for (int j = 0; j < 6; ++j) {
                am[j].x += v[j].x; am[j].y += v[j].y;
                am[j].z += v[j].z; am[j].w += v[j].w;
            }
            ++cm;
        }
    }

    #pragma unroll
    for (int j = 0; j < 6; ++j) {
        const int h = lane * 4 + 128 * j;
        atomicAdd(&s_p[h + 0], ap[j].x); atomicAdd(&s_p[h + 1], ap[j].y);
        atomicAdd(&s_p[h + 2], ap[j].z); atomicAdd(&s_p[h + 3], ap[j].w);
        atomicAdd(&s_m[h + 0], am[j].x); atomicAdd(&s_m[h + 1], am[j].y);
        atomicAdd(&s_m[h + 2], am[j].z); atomicAdd(&s_m[h + 3], am[j].w);
    }
    if (lane == 0) { atomicAdd(&s_cp, cp); atomicAdd(&s_cm, cm); }
    __syncthreads();

    for (int h = tid; h < HH; h += 256) {
        sum_p[(size_t)b * HH + h] = s_p[h];
        sum_m[(size_t)b * HH + h] = s_m[h];
    }
    if (tid == 0) { cnt_p[b] = (float)s_cp; cnt_m[b] = (float)s_cm; }
}

// ---------------------------------------------------------------------------
// Kernel 2: sums -> means in place, plus cnt>0 flags for the epilogue zeroing.
// ---------------------------------------------------------------------------
__global__ __launch_bounds__(256)
void normalize_kernel(float* __restrict__ sum_p, float* __restrict__ sum_m,
                      const float* __restrict__ cnt_p, const float* __restrict__ cnt_m,
                      float* __restrict__ flag_p, float* __restrict__ flag_m)
{
    const int b = blockIdx.x, tid = threadIdx.x;
    const float cp = cnt_p[b], cm = cnt_m[b];
    const float ip = cp > 0.f ? 1.f / cp : 0.f;
    const float im = cm > 0.f ? 1.f / cm : 0.f;
    for (int h = tid; h < HH; h += 256) {
        sum_p[(size_t)b * HH + h] *= ip;
        sum_m[(size_t)b * HH + h] *= im;
    }
    if (tid == 0) { flag_p[b] = cp > 0.f ? 1.f : 0.f; flag_m[b] = cm > 0.f ? 1.f : 0.f; }
}

// ---------------------------------------------------------------------------
// WMMA f32 GEMM: Out[M,N] = rowscale ∘ (A[M,K] @ W[N,K]^T + bias).
// One wave per 16x16 tile, K stepped by 4 via V_WMMA_F32_16X16X4_F32.
// A frag (16x4): lane L -> a = A[tm + L%16, k + 2*(L/16) + {0,1}]
// B frag (4x16): lane L -> b = W[tn + L%16, k + 2*(L/16) + {0,1}]   (W row-major [out,in])
// D (16x16):     lane L, reg r -> Out[tm + r + 8*(L/16), tn + L%16]
// ---------------------------------------------------------------------------
__global__ __launch_bounds__(256)
void gemm_wmma_kernel(const float* __restrict__ A, int lda,
                      const float* __restrict__ W, int ldw,
                      float* __restrict__ Out, int ldo,
                      const float* __restrict__ bias,
                      const float* __restrict__ rowscale,
                      int M, int N, int K)
{
    const int lane   = threadIdx.x & 31;
    const int waveId = blockIdx.x * (blockDim.x >> 5) + (threadIdx.x >> 5);
    const int ntN    = N >> 4;
    const int tm     = (waveId / ntN) << 4;
    const int tn     = (waveId % ntN) << 4;
    if (tm >= M) return;                       // wave-uniform

    const int half = lane >> 4;
    const int lr   = lane & 15;
    const float* arow = A + (size_t)(tm + lr) * lda + 2 * half;
    const float* wrow = W + (size_t)(tn + lr) * ldw + 2 * half;

    v8f acc = {0.f, 0.f, 0.f, 0.f, 0.f, 0.f, 0.f, 0.f};
    #pragma unroll 4
    for (int k = 0; k < K; k += 4) {
        v2f a = *(const v2f*)(arow + k);
        v2f w = *(const v2f*)(wrow + k);
        acc = __builtin_amdgcn_wmma_f32_16x16x4_f32(
            false, a, false, w, (short)0, acc, false, false);
    }

    const float bb = bias ? bias[tn + lr] : 0.f;
    #pragma unroll
    for (int r = 0; r < 8; ++r) {
        const int m = tm + r + 8 * half;
        float v = acc[r] + bb;
        if (rowscale) v *= rowscale[m];
        Out[(size_t)m * ldo + tn + lr] = v;
    }
}

// ---------------------------------------------------------------------------
// h_concat = [h+, h+*h-, h+ - h-, h-]   (256 x 3072)
// ---------------------------------------------------------------------------
__global__ __launch_bounds__(256)
void concat_kernel(const float* __restrict__ hp, const float* __restrict__ hm,
                   float* __restrict__ out)
{
    const int b = blockIdx.x, tid = threadIdx.x;
    const size_t base = (size_t)b * 4 * HH;
    for (int h = tid; h < HH; h += 256) {
        const float p = hp[(size_t)b * HH + h];
        const float m = hm[(size_t)b * HH + h];
        out[base + h]          = p;
        out[base + HH + h]     = p * m;
        out[base + 2 * HH + h] = p - m;
        out[base + 3 * HH + h] = m;
    }
}

// combined[:, 768:1536] = tweet_emb  (left half written by the W-comb GEMM)
__global__ __launch_bounds__(256)
void copy_tweet_kernel(const float* __restrict__ tweet, float* __restrict__ combined)
{
    const int b = blockIdx.x, tid = threadIdx.x;
    for (int h = tid; h < HH; h += 256)
        combined[(size_t)b * 1536 + HH + h] = tweet[(size_t)b * HH + h];
}

// ---------------------------------------------------------------------------
// Training-mode BatchNorm (biased var over the 256-row batch) + ReLU, in place.
// One thread per column; each thread owns its column fully, so no sync needed.
// ---------------------------------------------------------------------------
__global__ __launch_bounds__(256)
void bn_relu_kernel(float* __restrict__ X, int N,
                    const float* __restrict__ g, const float* __restrict__ beta)
{
    const int col = blockIdx.x * blockDim.x + threadIdx.x;
    if (col >= N) return;
    float s = 0.f, s2 = 0.f;
    for (int r = 0; r < BB; ++r) {
        const float v = X[(size_t)r * N + col];
        s += v; s2 += v * v;
    }
    const float mu  = s * (1.f / BB);
    const float var = s2 * (1.f / BB) - mu * mu;
    const float inv = rsqrtf(var + 1e-5f);
    const float gg = g[col], bt = beta[col];
    for (int r = 0; r < BB; ++r) {
        const float v = (X[(size_t)r * N + col] - mu) * inv * gg + bt;
        X[(size_t)r * N + col] = v > 0.f ? v : 0.f;
    }
}

// Final 256x384 @ [4,384]^T + b  -> 256x4
__global__ __launch_bounds__(256)
void final_kernel(const float* __restrict__ X, const float* __restrict__ w,
                  const float* __restrict__ b, float* __restrict__ out)
{
    const int idx = blockIdx.x * blockDim.x + threadIdx.x;   // 1024 threads
    const int m = idx >> 2, l = idx & 3;
    const float* xr = X + (size_t)m * 384;
    const float* wr = w + (size_t)l * 384;
    float s = b[l];
    for (int k = 0; k < 384; ++k) s += xr[k] * wr[k];
    out[idx] = s;
}

extern "C" void kernel_launch(void* const* d_in, const int* in_sizes, int n_in,
                              void* d_out, int out_size, void* d_ws, size_t ws_size,
                              hipStream_t stream)
{
    const float* tweet  = (const float*)d_in[0];
    const float* cemb   = (const float*)d_in[1];
    const float* wv_p   = (const float*)d_in[2];
    const float* bv_p   = (const float*)d_in[3];
    const float* wo_p   = (const float*)d_in[4];
    const float* bo_p   = (const float*)d_in[5];
    const float* wv_m   = (const float*)d_in[6];
    const float* bv_m   = (const float*)d_in[7];
    const float* wo_m   = (const float*)d_in[8];
    const float* bo_m   = (const float*)d_in[9];
    const float* w_comb = (const float*)d_in[10];
    const float* b_comb = (const float*)d_in[11];
    const float* c1_w   = (const float*)d_in[12];
    const float* c1_b   = (const float*)d_in[13];
    const float* bn1_g  = (const float*)d_in[14];
    const float* bn1_b  = (const float*)d_in[15];
    const float* c2_w   = (const float*)d_in[16];
    const float* c2_b   = (const float*)d_in[17];
    const float* bn2_g  = (const float*)d_in[18];
    const float* bn2_b  = (const float*)d_in[19];
    const float* c3_w   = (const float*)d_in[20];
    const float* c3_b   = (const float*)d_in[21];
    float* out = (float*)d_out;
    float* ws  = (float*)d_ws;

    const size_t BH = (size_t)BB * HH;   // 196608
    float* mean_p = ws;                  // sums -> means in place
    float* mean_m = mean_p + BH;
    float* cnt_p  = mean_m + BH;
    float* cnt_m  = cnt_p + BB;
    float* flag_p = cnt_m + BB;
    float* flag_m = flag_p + BB;
    float* tmp_p  = flag_m + BB;
    float* tmp_m  = tmp_p + BH;
    float* h_p    = tmp_m + BH;
    float* h_m    = h_p + BH;
    float* hcat   = h_m + BH;            // 256 x 3072
    float* comb   = hcat + (size_t)BB * 4 * HH;   // 256 x 1536
    float* y1     = comb + (size_t)BB * 2 * HH;   // 256 x 768
    float* y2     = y1 + BH;                      // 256 x 384

    // 1) stream comment_emb once: mask + masked sums (memory-bound floor ~7us)
    mask_sum_kernel<<<BB, 256, 0, stream>>>(tweet, cemb, mean_p, mean_m, cnt_p, cnt_m);
    normalize_kernel<<<BB, 256, 0, stream>>>(mean_p, mean_m, cnt_p, cnt_m, flag_p, flag_m);

    // 2) affine attention chains on the means (4 x [256,768]x[768,768])
    const int blk768 = (BB / 16) * (HH / 16) / 8;   // 96 blocks, 8 waves each
    const int blk384 = (BB / 16) * (384 / 16) / 8;  // 48 blocks
    gemm_wmma_kernel<<<blk768, 256, 0, stream>>>(mean_p, HH, wv_p, HH, tmp_p, HH,
                                                 bv_p, nullptr, BB, HH, HH);
    gemm_wmma_kernel<<<blk768, 256, 0, stream>>>(tmp_p, HH, wo_p, HH, h_p, HH,
                                                 bo_p, flag_p, BB, HH, HH);
    gemm_wmma_kernel<<<blk768, 256, 0, stream>>>(mean_m, HH, wv_m, HH, tmp_m, HH,
                                                 bv_m, nullptr, BB, HH, HH);
    gemm_wmma_kernel<<<blk768, 256, 0, stream>>>(tmp_m, HH, wo_m, HH, h_m, HH,
                                                 bo_m, flag_m, BB, HH, HH);

    // 3) h_concat, W-comb GEMM (writes left half of `combined`), tweet copy
    concat_kernel<<<BB, 256, 0, stream>>>(h_p, h_m, hcat);
    gemm_wmma_kernel<<<blk768, 256, 0, stream>>>(hcat, 4 * HH, w_comb, 4 * HH,
                                                 comb, 2 * HH, b_comb, nullptr,
                                                 BB, HH, 4 * HH);
    copy_tweet_kernel<<<BB, 256, 0, stream>>>(tweet, comb);

    // 4) classifier
    gemm_wmma_kernel<<<blk768, 256, 0, stream>>>(comb, 2 * HH, c1_w, 2 * HH,
                                                 y1, HH, c1_b, nullptr, BB, HH, 2 * HH);
    bn_relu_kernel<<<(HH + 255) / 256, 256, 0, stream>>>(y1, HH, bn1_g, bn1_b);
    gemm_wmma_kernel<<<blk384, 256, 0, stream>>>(y1, HH, c2_w, HH,
                                                 y2, 384, c2_b, nullptr, BB, 384, HH);
    bn_relu_kernel<<<(384 + 255) / 256, 256, 0, stream>>>(y2, 384, bn2_g, bn2_b);
    final_kernel<<<4, 256, 0, stream>>>(y2, c3_w, c3_b, out);
}